// GATClassifier_12197707121343
// MI455X (gfx1250) — compile-verified
//
#include <hip/hip_runtime.h>
#include <math.h>

// ---------------------------------------------------------------------------
// CDNA5 / gfx1250 GAT + LSTM pipeline.
// GEMMs: V_WMMA_F32_16X16X4_F32 (exact f32), A tile staged to LDS via
// GLOBAL_LOAD_ASYNC_TO_LDS_B128 (+ s_wait_asynccnt) when available.
// ---------------------------------------------------------------------------

typedef float v2f __attribute__((ext_vector_type(2)));
typedef float v4f __attribute__((ext_vector_type(4)));
typedef float v8f __attribute__((ext_vector_type(8)));
typedef int   v4i __attribute__((ext_vector_type(4)));

#if defined(__AMDGCN__) && defined(__has_builtin)
#if __has_builtin(__builtin_amdgcn_global_load_async_to_lds_b128) && \
    __has_builtin(__builtin_amdgcn_s_wait_asynccnt)
#define USE_ASYNC_LDS 1
#endif
#endif

// ---------------- WMMA f32 GEMM:  C[M,N] = A[M,K] @ B[N,K]^T (+bias0+bias1) --
// One block -> one 16-row tile of C; wave w -> column tile w.
// Requires M%16==0, N%16==0, K%4==0, K<=256 (true for all uses here).
__global__ __launch_bounds__(256)
void wmma_gemm_f32(const float* __restrict__ A, const float* __restrict__ B,
                   const float* __restrict__ bias0, const float* __restrict__ bias1,
                   float* __restrict__ C, int M, int N, int K)
{
    __shared__ float tileA[16 * (256 + 4)];     // padded rows: stride K+4 (bank-conflict-free)
    const int KP   = K + 4;
    const int mt   = blockIdx.x;
    const int lane = threadIdx.x & 31;
    const int wid  = threadIdx.x >> 5;
    const int nwv  = blockDim.x >> 5;
    const int ntN  = N >> 4;

    // ---- stage A tile (16 x K) into LDS: 16B chunks, async DMA when available
    {
        const int chunksPerRow = K >> 2;                   // float4 chunks
        const int nchunk = 16 * chunksPerRow;
        const float* Abase = A + (size_t)mt * 16 * K;      // 16 consecutive rows: contiguous
        for (int c = threadIdx.x; c < nchunk; c += blockDim.x) {
            const int r = c / chunksPerRow, j = c % chunksPerRow;
            const float* src = Abase + r * K + j * 4;
            float*       dst = tileA + r * KP + j * 4;
#ifdef USE_ASYNC_LDS
            __builtin_amdgcn_global_load_async_to_lds_b128(
                (__attribute__((address_space(1))) v4i*)src,
                (__attribute__((address_space(3))) v4i*)dst, 0, 0);
#else
            *(v4f*)dst = *(const v4f*)src;
#endif
        }
#ifdef USE_ASYNC_LDS
        __builtin_amdgcn_s_wait_asynccnt(0);               // LDS writes landed
#endif
        __syncthreads();
    }

    const int hi = lane >> 4;                  // half-wave select
    const int l  = lane & 15;

    // A 16x4 fragment from LDS: lane l holds {A[l][k+2*hi], A[l][k+1+2*hi]}
    const float* Afrag = tileA + l * KP + hi * 2;

    for (int nt = wid; nt < ntN; nt += nwv) {
        // B 4x16 fragment (B[k][n] = W[n][k]): lane l holds {W[col][k+2*hi], W[col][k+1+2*hi]}
        const float* Brow = B + (size_t)(nt * 16 + l) * K + hi * 2;

        v8f acc = {};
        for (int k0 = 0; k0 < K; k0 += 4) {
            v2f a = *(const v2f*)(Afrag + k0);
            v2f b = *(const v2f*)(Brow + k0);
            acc = __builtin_amdgcn_wmma_f32_16x16x4_f32(
                /*neg_a=*/false, a, /*neg_b=*/false, b,
                /*c_mod=*/(short)0, acc, /*reuse_a=*/false, /*reuse_b=*/false);
        }

        const int col = nt * 16 + l;
        float bsum = 0.f;
        if (bias0) bsum += bias0[col];
        if (bias1) bsum += bias1[col];
#pragma unroll
        for (int r = 0; r < 8; ++r) {
            int row = mt * 16 + r + hi * 8;    // C: acc[r] = row r (lo half) / r+8 (hi half)
            C[(size_t)row * N + col] = acc[r] + bsum;
        }
    }
}

// ---------------- per-node attention logits: al[n,h] = sum_c h[n,h,c]*a[h,c] --
__global__ void alpha_kernel(const float* __restrict__ h,
                             const float* __restrict__ aS, const float* __restrict__ aD,
                             float* __restrict__ alS, float* __restrict__ alD,
                             int Nn, int H, int C)
{
    const int w    = (int)((blockIdx.x * (long long)blockDim.x + threadIdx.x) >> 5);
    const int lane = threadIdx.x & 31;
    if (w >= Nn) return;
    const float* hrow = h + (size_t)w * H * C;
    for (int hh = 0; hh < H; ++hh) {
        float s = 0.f, d = 0.f;
        for (int c = lane; c < C; c += 32) {
            float v = hrow[hh * C + c];
            s += v * aS[hh * C + c];
            d += v * aD[hh * C + c];
        }
        for (int off = 16; off; off >>= 1) {
            s += __shfl_xor(s, off, 32);
            d += __shfl_xor(d, off, 32);
        }
        if (lane == 0) { alS[w * H + hh] = s; alD[w * H + hh] = d; }
    }
}

__global__ void fill_kernel(float* __restrict__ p, long long n, float v)
{
    long long i = blockIdx.x * (long long)blockDim.x + threadIdx.x;
    if (i < n) p[i] = v;
}

// ordered-int trick: monotone float max via signed max (>=0) / unsigned min (<0)
__device__ inline void atomicMaxF(float* addr, float v)
{
    if (v >= 0.f) atomicMax((int*)addr, __float_as_int(v));
    else          atomicMin((unsigned int*)addr, __float_as_uint(v));
}

__device__ inline void edge_sd(const long long* __restrict__ ei, long long Ein,
                               long long e, long long& s, long long& d)
{
    if (e < Ein) { s = ei[e]; d = ei[Ein + e]; }
    else         { s = e - Ein; d = s; }      // appended self loops
}

__device__ inline float lrelu(float v) { return v > 0.f ? v : 0.2f * v; }

// ---------------- pass A: segment max over incoming edges ------------------
__global__ void edge_max_kernel(const long long* __restrict__ ei, long long Ein, long long Etot,
                                const float* __restrict__ alS, const float* __restrict__ alD,
                                float* __restrict__ m, int H)
{
    long long idx = blockIdx.x * (long long)blockDim.x + threadIdx.x;
    if (idx >= Etot * H) return;
    long long e = idx / H; int hh = (int)(idx % H);
    long long s, d; edge_sd(ei, Ein, e, s, d);
    float v = lrelu(alS[s * H + hh] + alD[d * H + hh]);
    atomicMaxF(&m[d * H + hh], v);
}

// ---------------- pass B: denom = segment_sum(exp(e - m)) ------------------
__global__ void edge_sum_kernel(const long long* __restrict__ ei, long long Ein, long long Etot,
                                const float* __restrict__ alS, const float* __restrict__ alD,
                                const float* __restrict__ m, float* __restrict__ den, int H)
{
    long long idx = blockIdx.x * (long long)blockDim.x + threadIdx.x;
    if (idx >= Etot * H) return;
    long long e = idx / H; int hh = (int)(idx % H);
    long long s, d; edge_sd(ei, Ein, e, s, d);
    float v = lrelu(alS[s * H + hh] + alD[d * H + hh]);
    unsafeAtomicAdd(&den[d * H + hh], __expf(v - m[d * H + hh]));
}

// ---------------- pass C: out[dst] += alpha * h[src] (wave per edge) -------
__global__ void edge_scatter_kernel(const long long* __restrict__ ei, long long Ein, long long Etot,
                                    const float* __restrict__ alS, const float* __restrict__ alD,
                                    const float* __restrict__ m, const float* __restrict__ den,
                                    const float* __restrict__ hsrc, float* __restrict__ out,
                                    int H, int C)
{
    long long w = (blockIdx.x * (long long)blockDim.x + threadIdx.x) >> 5;
    const int lane = threadIdx.x & 31;
    if (w >= Etot) return;
    long long s, d; edge_sd(ei, Ein, w, s, d);
    const int F = H * C;
    for (int hh = 0; hh < H; ++hh) {
        float v = lrelu(alS[s * H + hh] + alD[d * H + hh]);
        float alpha = __expf(v - m[d * H + hh]) / den[d * H + hh];
        for (int c = lane; c < C; c += 32)
            unsafeAtomicAdd(&out[d * (size_t)F + hh * C + c],
                            alpha * hsrc[s * (size_t)F + hh * C + c]);
    }
}

// ---------------- epilogue: buf = (buf + bias)  [optional elu] -------------
__global__ void finalize_kernel(float* __restrict__ buf, const float* __restrict__ bias,
                                long long total, int F, int do_elu)
{
    long long i = blockIdx.x * (long long)blockDim.x + threadIdx.x;
    if (i >= total) return;
    float v = buf[i] + bias[(int)(i % F)];
    if (do_elu) v = v > 0.f ? v : expm1f(v);
    buf[i] = v;
}

// ---------------- sequential LSTM + FC, single wave32 ----------------------
// lane l owns hidden unit l; W_hh kept in registers; h broadcast via shfl.
__global__ __launch_bounds__(32)
void lstm_fc_kernel(const float* __restrict__ pre, const float* __restrict__ whh,
                    const float* __restrict__ fcw, const float* __restrict__ fcb,
                    float* __restrict__ out, int Nn)
{
    const int lane = threadIdx.x;               // 0..31
    float w[4][32];                             // W_hh[(g*32+lane)][j]
#pragma unroll
    for (int g = 0; g < 4; ++g)
#pragma unroll
        for (int j = 0; j < 32; ++j)
            w[g][j] = whh[(g * 32 + lane) * 32 + j];
    const float fw   = fcw[lane];
    const float fcb0 = fcb[0];

    float hp = 0.f, cp = 0.f;
    for (int n = 0; n < Nn; ++n) {
        const float* p = pre + (size_t)n * 128;
        float a0 = p[lane], a1 = p[32 + lane], a2 = p[64 + lane], a3 = p[96 + lane];
#pragma unroll
        for (int j = 0; j < 32; ++j) {
            float hj = __shfl(hp, j, 32);
            a0 += hj * w[0][j];
            a1 += hj * w[1][j];
            a2 += hj * w[2][j];
            a3 += hj * w[3][j];
        }
        float ig = 1.f / (1.f + __expf(-a0));
        float fg = 1.f / (1.f + __expf(-a1));
        float gg = tanhf(a2);
        float og = 1.f / (1.f + __expf(-a3));
        float c  = fg * cp + ig * gg;
        float hn = og * tanhf(c);

        float y = hn * fw;
        for (int off = 16; off; off >>= 1) y += __shfl_xor(y, off, 32);
        if (lane == 0) out[n] = y + fcb0;

        hp = hn; cp = c;
    }
}

// ---------------------------------------------------------------------------
extern "C" void kernel_launch(void* const* d_in, const int* in_sizes, int n_in,
                              void* d_out, int out_size, void* d_ws, size_t ws_size,
                              hipStream_t stream)
{
    const float* x   = (const float*)d_in[0];
    const float* w1  = (const float*)d_in[1];
    const float* a1s = (const float*)d_in[2];
    const float* a1d = (const float*)d_in[3];
    const float* b1  = (const float*)d_in[4];
    const float* w2  = (const float*)d_in[5];
    const float* a2s = (const float*)d_in[6];
    const float* a2d = (const float*)d_in[7];
    const float* b2  = (const float*)d_in[8];
    const float* wih = (const float*)d_in[9];
    const float* whh = (const float*)d_in[10];
    const float* bih = (const float*)d_in[11];
    const float* bhh = (const float*)d_in[12];
    const float* fcw = (const float*)d_in[13];
    const float* fcb = (const float*)d_in[14];
    const long long* ei = (const long long*)d_in[15];

    const int IN = 256, F1 = 128, H1 = 4, C1 = 32, F2 = 64, H2 = 1, C2 = 64, G4 = 128;
    const int Nn = in_sizes[0] / IN;               // 20000
    const long long Ein  = in_sizes[15] / 2;       // 640000
    const long long Etot = Ein + Nn;               // + self loops

    // -------- workspace layout (floats) --------
    float* p    = (float*)d_ws;
    float* h1   = p; p += (size_t)Nn * F1;
    float* alS1 = p; p += (size_t)Nn * H1;
    float* alD1 = p; p += (size_t)Nn * H1;
    float* m1   = p; p += (size_t)Nn * H1;
    float* den1 = p; p += (size_t)Nn * H1;
    float* out1 = p; p += (size_t)Nn * F1;         // -> x2 after bias+elu
    float* h2   = p; p += (size_t)Nn * F2;
    float* alS2 = p; p += (size_t)Nn * H2;
    float* alD2 = p; p += (size_t)Nn * H2;
    float* m2   = p; p += (size_t)Nn * H2;
    float* den2 = p; p += (size_t)Nn * H2;
    float* out2 = p; p += (size_t)Nn * F2;
    float* preb = p; p += (size_t)Nn * G4;

    const int TB = 256;
    auto cdiv = [](long long a, long long b) { return (unsigned int)((a + b - 1) / b); };

    // ================= layer 1: GAT(256 -> 4x32) =================
    wmma_gemm_f32<<<Nn / 16, (F1 / 16) * 32, 0, stream>>>(x, w1, nullptr, nullptr, h1, Nn, F1, IN);
    alpha_kernel<<<cdiv((long long)Nn * 32, TB), TB, 0, stream>>>(h1, a1s, a1d, alS1, alD1, Nn, H1, C1);

    fill_kernel<<<cdiv((long long)Nn * H1, TB), TB, 0, stream>>>(m1, (long long)Nn * H1, -INFINITY);
    (void)hipMemsetAsync(den1, 0, (size_t)Nn * H1 * sizeof(float), stream);
    (void)hipMemsetAsync(out1, 0, (size_t)Nn * F1 * sizeof(float), stream);

    edge_max_kernel<<<cdiv(Etot * H1, TB), TB, 0, stream>>>(ei, Ein, Etot, alS1, alD1, m1, H1);
    edge_sum_kernel<<<cdiv(Etot * H1, TB), TB, 0, stream>>>(ei, Ein, Etot, alS1, alD1, m1, den1, H1);
    edge_scatter_kernel<<<cdiv(Etot * 32, TB), TB, 0, stream>>>(ei, Ein, Etot, alS1, alD1, m1, den1,
                                                                h1, out1, H1, C1);
    finalize_kernel<<<cdiv((long long)Nn * F1, TB), TB, 0, stream>>>(out1, b1, (long long)Nn * F1, F1, 1);

    // ================= layer 2: GAT(128 -> 64, single head) =================
    wmma_gemm_f32<<<Nn / 16, (F2 / 16) * 32, 0, stream>>>(out1, w2, nullptr, nullptr, h2, Nn, F2, F1);
    alpha_kernel<<<cdiv((long long)Nn * 32, TB), TB, 0, stream>>>(h2, a2s, a2d, alS2, alD2, Nn, H2, C2);

    fill_kernel<<<cdiv((long long)Nn * H2, TB), TB, 0, stream>>>(m2, (long long)Nn * H2, -INFINITY);
    (void)hipMemsetAsync(den2, 0, (size_t)Nn * H2 * sizeof(float), stream);
    (void)hipMemsetAsync(out2, 0, (size_t)Nn * F2 * sizeof(float), stream);

    edge_max_kernel<<<cdiv(Etot * H2, TB), TB, 0, stream>>>(ei, Ein, Etot, alS2, alD2, m2, H2);
    edge_sum_kernel<<<cdiv(Etot * H2, TB), TB, 0, stream>>>(ei, Ein, Etot, alS2, alD2, m2, den2, H2);
    edge_scatter_kernel<<<cdiv(Etot * 32, TB), TB, 0, stream>>>(ei, Ein, Etot, alS2, alD2, m2, den2,
                                                                h2, out2, H2, C2);
    finalize_kernel<<<cdiv((long long)Nn * F2, TB), TB, 0, stream>>>(out2, b2, (long long)Nn * F2, F2, 0);

    // ================= LSTM input projection: pre = out2 @ Wih^T + bih + bhh ===
    wmma_gemm_f32<<<Nn / 16, (G4 / 16) * 32, 0, stream>>>(out2, wih, bih, bhh, preb, Nn, G4, F2);

    // ================= sequential LSTM + FC, single wave32 =================
    lstm_fc_kernel<<<1, 32, 0, stream>>>(preb, whh, fcw, fcb, (float*)d_out, Nn);
}